// AttnIO_11854109737008
// MI455X (gfx1250) — compile-verified
//
#include <hip/hip_runtime.h>
#include <math.h>

// Problem constants (from the reference)
#define NN    20000     // nodes
#define EE    320000    // edges
#define HH    4         // heads
#define NREL  50
#define NSEED 32
#define SLOPE 0.01f

typedef __attribute__((ext_vector_type(2))) float v2f;
typedef __attribute__((ext_vector_type(8))) float v8f;

// ---- order-preserving float<->uint encoding for atomicMax on floats ----
__device__ __forceinline__ unsigned f2ord(float f) {
  unsigned u = __float_as_uint(f);
  return (u & 0x80000000u) ? ~u : (u | 0x80000000u);
}
__device__ __forceinline__ float ord2f(unsigned u) {
  return (u & 0x80000000u) ? __uint_as_float(u & 0x7FFFFFFFu) : __uint_as_float(~u);
}

// =====================================================================
// GEMM: C[M x 64] = A[M x K] @ W[K x 64] (+ bias[64]); optional row gather.
// One wave (32 lanes) per 16-row tile, 4 accumulators of 16x16 f32.
// Uses V_WMMA_F32_16X16X4_F32 (full f32; workload is bandwidth-bound so
// there is no reason to drop precision to bf16/fp8).
// =====================================================================
template <bool GATHER>
__global__ void gemm_nx64_wmma(const float* __restrict__ A, const float* __restrict__ W,
                               const int* __restrict__ idx, const float* __restrict__ bias,
                               float* __restrict__ C, int M, int K) {
  const int lane  = threadIdx.x;          // blockDim.x == 32
  const int mloc  = lane & 15;            // row within tile (A) / col (B,C)
  const int khalf = (lane >> 4) << 1;     // 0 or 2 : K offset of this lane-half
  const int row0  = blockIdx.x << 4;

  int arow = row0 + mloc;
  int asrc = 0;
  if (arow < M) asrc = GATHER ? idx[arow] : arow;
  const float* Ar = A + (size_t)asrc * K;

  v8f c0 = {}, c1 = {}, c2 = {}, c3 = {};
  for (int k0 = 0; k0 < K; k0 += 4) {
    v2f a;
    a.x = Ar[k0 + khalf];
    a.y = Ar[k0 + khalf + 1];
    const float* W0 = W + (size_t)(k0 + khalf) * 64 + mloc;
    v2f b0, b1, b2, b3;
    b0.x = W0[0];   b0.y = W0[64];
    b1.x = W0[16];  b1.y = W0[80];
    b2.x = W0[32];  b2.y = W0[96];
    b3.x = W0[48];  b3.y = W0[112];
    c0 = __builtin_amdgcn_wmma_f32_16x16x4_f32(false, a, false, b0, (short)0, c0, false, false);
    c1 = __builtin_amdgcn_wmma_f32_16x16x4_f32(false, a, false, b1, (short)0, c1, false, false);
    c2 = __builtin_amdgcn_wmma_f32_16x16x4_f32(false, a, false, b2, (short)0, c2, false, false);
    c3 = __builtin_amdgcn_wmma_f32_16x16x4_f32(false, a, false, b3, (short)0, c3, false, false);
  }

  float bb0 = 0.f, bb1 = 0.f, bb2 = 0.f, bb3 = 0.f;
  if (bias) { bb0 = bias[mloc]; bb1 = bias[mloc + 16]; bb2 = bias[mloc + 32]; bb3 = bias[mloc + 48]; }

  const int mhi = (lane >> 4) << 3;       // C/D: VGPR r -> row r (+8 for upper lanes)
#pragma unroll
  for (int r = 0; r < 8; ++r) {
    int m = row0 + mhi + r;
    if (m < M) {
      float* Cr = C + (size_t)m * 64 + mloc;
      Cr[0]  = c0[r] + bb0;
      Cr[16] = c1[r] + bb1;
      Cr[32] = c2[r] + bb2;
      Cr[48] = c3[r] + bb3;
    }
  }
}

// out[64] = x[0:64] @ W[64x64]  (tiny: dialogue-context projections)
__global__ void k_gemv64(const float* __restrict__ x, const float* __restrict__ W,
                         float* __restrict__ out) {
  int j = threadIdx.x;                    // 64 threads
  float acc = 0.f;
#pragma unroll
  for (int k = 0; k < 64; ++k) acc += x[k] * W[k * 64 + j];
  out[j] = acc;
}

// ---- inflow: e[e,h] = leakyrelu( dot(fs[src]+relp[et], edst[h][dst]) ), seg-max by dst
__global__ void k_edge_in(const float* __restrict__ fs, const float* __restrict__ edst,
                          const float* __restrict__ relp,
                          const int* __restrict__ src, const int* __restrict__ dst,
                          const int* __restrict__ et,
                          float* __restrict__ z, unsigned* __restrict__ mmax) {
  int e = (blockIdx.x * blockDim.x + threadIdx.x) >> 5;   // one wave per edge
  int lane = threadIdx.x & 31;
  if (e >= EE) return;
  int s = src[e], d = dst[e], t = et[e];
  float p0 = fs[(size_t)s * 64 + lane]      + relp[t * 64 + lane];
  float p1 = fs[(size_t)s * 64 + lane + 32] + relp[t * 64 + lane + 32];
#pragma unroll
  for (int h = 0; h < HH; ++h) {
    const float* ed = edst + ((size_t)h * NN + d) * 64;
    float v = p0 * ed[lane] + p1 * ed[lane + 32];
#pragma unroll
    for (int off = 16; off > 0; off >>= 1) v += __shfl_down(v, off, 32);
    if (lane == 0) {
      float ee = v > 0.f ? v : v * SLOPE;
      z[(size_t)e * HH + h] = ee;
      atomicMax(&mmax[d * HH + h], f2ord(ee));
    }
  }
}

// ---- outflow: e[e,h] = leakyrelu( dot(ef[dst], esrc[h][src]) + dot(relp[et], ef[src]) ), seg-max by src
__global__ void k_edge_out(const float* __restrict__ ef, const float* __restrict__ esrc,
                           const float* __restrict__ relp,
                           const int* __restrict__ src, const int* __restrict__ dst,
                           const int* __restrict__ et,
                           float* __restrict__ z, unsigned* __restrict__ mmax) {
  int e = (blockIdx.x * blockDim.x + threadIdx.x) >> 5;
  int lane = threadIdx.x & 31;
  if (e >= EE) return;
  int s = src[e], d = dst[e], t = et[e];
  const float* efs = ef + (size_t)s * 64;
  const float* efd = ef + (size_t)d * 64;
  float q = relp[t * 64 + lane] * efs[lane] + relp[t * 64 + lane + 32] * efs[lane + 32];
#pragma unroll
  for (int off = 16; off > 0; off >>= 1) q += __shfl_down(q, off, 32);
#pragma unroll
  for (int h = 0; h < HH; ++h) {
    const float* pr = esrc + ((size_t)h * NN + s) * 64;
    float v = efd[lane] * pr[lane] + efd[lane + 32] * pr[lane + 32];
#pragma unroll
    for (int off = 16; off > 0; off >>= 1) v += __shfl_down(v, off, 32);
    if (lane == 0) {
      float ee = v + q;
      ee = ee > 0.f ? ee : ee * SLOPE;
      z[(size_t)e * HH + h] = ee;
      atomicMax(&mmax[s * HH + h], f2ord(ee));
    }
  }
}

// z <- exp(z - max[seg]);  ssum[seg] += z
__global__ void k_expsum(float* __restrict__ z, const unsigned* __restrict__ mmax,
                         float* __restrict__ ssum, const int* __restrict__ seg) {
  int t = blockIdx.x * blockDim.x + threadIdx.x;
  if (t >= EE * HH) return;
  int e = t >> 2, h = t & 3;
  int g = seg[e];
  float zz = expf(z[t] - ord2f(mmax[g * HH + h]));
  z[t] = zz;
  atomicAdd(&ssum[g * HH + h], zz);
}

// rst[dst, h*64+d] += (fs[src,d]+relp[et,d]) * z[e,h]/ssum[dst,h]
__global__ void k_msg(const float* __restrict__ fs, const float* __restrict__ relp,
                      const float* __restrict__ z, const float* __restrict__ ssum,
                      const int* __restrict__ src, const int* __restrict__ dst,
                      const int* __restrict__ et, float* __restrict__ rst) {
  int t = blockIdx.x * blockDim.x + threadIdx.x;
  int e = t >> 6, dd = t & 63;
  if (e >= EE) return;
  int s = src[e], d = dst[e], ty = et[e];
  float p = fs[(size_t)s * 64 + dd] + relp[ty * 64 + dd];
  float* out = rst + (size_t)d * 256 + dd;
#pragma unroll
  for (int h = 0; h < HH; ++h)
    atomicAdd(out + h * 64, p * (z[(size_t)e * HH + h] / ssum[d * HH + h]));
}

// ---- outflow scoring / softmax over N ----
__global__ void k_score(const float* __restrict__ ef0, const float* __restrict__ dctx,
                        float* __restrict__ score) {
  int n = blockIdx.x * blockDim.x + threadIdx.x;
  if (n >= NN) return;
  const float* r = ef0 + (size_t)n * 64;
  float acc = 0.f;
#pragma unroll
  for (int d = 0; d < 64; ++d) acc += r[d] * dctx[d];
  score[n] = acc - 10000.f;
}
__global__ void k_seed(float* __restrict__ score, const int* __restrict__ seed) {
  int t = threadIdx.x;
  if (t < NSEED) score[seed[t]] += 10000.f;
}
__global__ void k_max(const float* __restrict__ score, unsigned* __restrict__ redm) {
  int n = blockIdx.x * blockDim.x + threadIdx.x;
  if (n < NN) atomicMax(redm, f2ord(score[n]));
}
__global__ void k_sumexp(const float* __restrict__ score, const unsigned* __restrict__ redm,
                         float* __restrict__ reds) {
  int n = blockIdx.x * blockDim.x + threadIdx.x;
  if (n < NN) atomicAdd(reds, expf(score[n] - ord2f(*redm)));
}
__global__ void k_norm(const float* __restrict__ score, const unsigned* __restrict__ redm,
                       const float* __restrict__ reds, float* __restrict__ a) {
  int n = blockIdx.x * blockDim.x + threadIdx.x;
  if (n < NN) a[n] = expf(score[n] - ord2f(*redm)) / (*reds);
}

// a_out[dst] += a_in[src] * mean_h( z[e,h]/ssum[src,h] )
__global__ void k_prop(const int* __restrict__ src, const int* __restrict__ dst,
                       const float* __restrict__ z, const float* __restrict__ ssum,
                       const float* __restrict__ a_in, float* __restrict__ a_out) {
  int e = blockIdx.x * blockDim.x + threadIdx.x;
  if (e >= EE) return;
  int s = src[e], d = dst[e];
  float t = 0.f;
#pragma unroll
  for (int h = 0; h < HH; ++h) t += z[(size_t)e * HH + h] / ssum[s * HH + h];
  atomicAdd(&a_out[d], a_in[s] * (t * (1.f / HH)));
}

// =====================================================================
extern "C" void kernel_launch(void* const* d_in, const int* in_sizes, int n_in,
                              void* d_out, int out_size, void* d_ws, size_t ws_size,
                              hipStream_t stream) {
  (void)in_sizes; (void)n_in; (void)out_size; (void)ws_size;
  const int*   src  = (const int*)d_in[0];
  const int*   dst  = (const int*)d_in[1];
  const int*   et   = (const int*)d_in[2];
  const int*   nid  = (const int*)d_in[3];
  const int*   seed = (const int*)d_in[4];
  const float* dlg  = (const float*)d_in[5];   // (1,64)
  const float* ent  = (const float*)d_in[6];   // (100000,64)
  const float* rel  = (const float*)d_in[7];   // (50,64)
  const float* fcw  = (const float*)d_in[8];   // (64,64)
  const float* wq   = (const float*)d_in[9];   // (4,64,64)
  const float* whe  = (const float*)d_in[10];  // (256,64)
  const float* whd  = (const float*)d_in[11];  // (64,64)
  const float* owi  = (const float*)d_in[12];  // (64,64)
  const float* owq  = (const float*)d_in[13];  // (4,64,64)

  char* ws = (char*)d_ws;
  size_t off = 0;
  auto alloc = [&](size_t bytes) -> void* {
    void* p = ws + off;
    off += (bytes + 255) & ~(size_t)255;
    return p;
  };
  float*    feat  = (float*)alloc((size_t)NN * 64 * 4);
  float*    ef0   = (float*)alloc((size_t)NN * 64 * 4);
  float*    ef1   = (float*)alloc((size_t)NN * 64 * 4);
  float*    ef2   = (float*)alloc((size_t)NN * 64 * 4);
  float*    relp  = (float*)alloc((size_t)NREL * 64 * 4);
  float*    edst  = (float*)alloc((size_t)HH * NN * 64 * 4);   // also reused as esrc
  float*    z     = (float*)alloc((size_t)EE * HH * 4);
  unsigned* mmax  = (unsigned*)alloc((size_t)NN * HH * 4);
  float*    ssum  = (float*)alloc((size_t)NN * HH * 4);
  float*    rst   = (float*)alloc((size_t)NN * 256 * 4);
  float*    hbias = (float*)alloc(64 * 4);
  float*    dctx  = (float*)alloc(64 * 4);
  float*    score = (float*)alloc((size_t)NN * 4);
  unsigned* redm  = (unsigned*)alloc(4);
  float*    reds  = (float*)alloc(4);
  float*    a0    = (float*)alloc((size_t)NN * 4);
  float*    a1    = (float*)alloc((size_t)NN * 4);

  dim3 b32(32), b64(64), b256(256);
  dim3 gGemm((NN + 15) / 16), gEdge(EE / 8), gEH((EE * HH + 255) / 256);
  dim3 gN((NN + 255) / 256);

  // dense prologue
  gemm_nx64_wmma<false><<<dim3((NREL + 15) / 16), b32, 0, stream>>>(rel, fcw, nullptr, nullptr, relp, NREL, 64);
  k_gemv64<<<dim3(1), b64, 0, stream>>>(dlg, whd, hbias);
  k_gemv64<<<dim3(1), b64, 0, stream>>>(dlg, owi, dctx);
  gemm_nx64_wmma<true><<<gGemm, b32, 0, stream>>>(ent, fcw, nid, nullptr, feat, NN, 64);

  // 3 inflow layers
  const float* fs_l[3] = {feat, ef0, ef1};
  float*       ef_l[3] = {ef0, ef1, ef2};
  for (int l = 0; l < 3; ++l) {
    for (int h = 0; h < HH; ++h)
      gemm_nx64_wmma<false><<<gGemm, b32, 0, stream>>>(fs_l[l], wq + (size_t)h * 64 * 64, nullptr,
                                                       nullptr, edst + (size_t)h * NN * 64, NN, 64);
    (void)hipMemsetAsync(mmax, 0, (size_t)NN * HH * 4, stream);
    (void)hipMemsetAsync(ssum, 0, (size_t)NN * HH * 4, stream);
    (void)hipMemsetAsync(rst, 0, (size_t)NN * 256 * 4, stream);
    k_edge_in<<<gEdge, b256, 0, stream>>>(fs_l[l], edst, relp, src, dst, et, z, mmax);
    k_expsum<<<gEH, b256, 0, stream>>>(z, mmax, ssum, dst);
    k_msg<<<dim3((EE * 64) / 256), b256, 0, stream>>>(fs_l[l], relp, z, ssum, src, dst, et, rst);
    gemm_nx64_wmma<false><<<gGemm, b32, 0, stream>>>(rst, whe, nullptr, hbias, ef_l[l], NN, 256);
  }

  // outflow: seed-biased softmax over nodes
  k_score<<<gN, b256, 0, stream>>>(ef0, dctx, score);
  k_seed<<<dim3(1), b32, 0, stream>>>(score, seed);
  (void)hipMemsetAsync(redm, 0, 4, stream);
  (void)hipMemsetAsync(reds, 0, 4, stream);
  k_max<<<gN, b256, 0, stream>>>(score, redm);
  k_sumexp<<<gN, b256, 0, stream>>>(score, redm, reds);
  k_norm<<<gN, b256, 0, stream>>>(score, redm, reds, a0);

  // 2 propagation steps; last one writes d_out directly
  const float* efo[2] = {ef1, ef2};
  float* a_in = a0;
  for (int it = 0; it < 2; ++it) {
    float* a_out = (it == 0) ? a1 : (float*)d_out;
    for (int h = 0; h < HH; ++h)
      gemm_nx64_wmma<false><<<gGemm, b32, 0, stream>>>(efo[it], owq + (size_t)h * 64 * 64, nullptr,
                                                       nullptr, edst + (size_t)h * NN * 64, NN, 64);
    (void)hipMemsetAsync(mmax, 0, (size_t)NN * HH * 4, stream);
    (void)hipMemsetAsync(ssum, 0, (size_t)NN * HH * 4, stream);
    (void)hipMemsetAsync(a_out, 0, (size_t)NN * 4, stream);
    k_edge_out<<<gEdge, b256, 0, stream>>>(efo[it], edst, relp, src, dst, et, z, mmax);
    k_expsum<<<gEH, b256, 0, stream>>>(z, mmax, ssum, src);
    k_prop<<<dim3((EE + 255) / 256), b256, 0, stream>>>(src, dst, z, ssum, a_in, a_out);
    a_in = a_out;
  }
}